// GeneralHeteroConv_36550171689596
// MI455X (gfx1250) — compile-verified
//
#include <hip/hip_runtime.h>
#include <hip/hip_bf16.h>

// ---------------------------------------------------------------------------
// EdgeConv restructuring:
//   msg = relu([x_i, x_j - x_i] @ W^T + b)
//       = relu( (x_i @ (Wa - Wb)^T) + (x_j @ Wb^T) + b )
// Precompute P = X(Wa-Wb)^T, Q = X Wb^T per edge type (WMMA f32 GEMM),
// then the per-edge stage is gather + add + relu + atomic aggregate.
// Max pass (atomic u32 max of nonneg floats, out init 0) runs first,
// sum pass atomicAdd's on top -> fused out_tp + out_int.
// ---------------------------------------------------------------------------

#define N_NODES 50000
#define CCH     256          // channels (IN == OUT == 256)
#define LDS_STRIDE 260       // 256 + 4 pad -> conflict-free fragment reads

typedef float v2f __attribute__((ext_vector_type(2)));
typedef float v8f __attribute__((ext_vector_type(8)));

// ---------------------------------------------------------------------------
// Pack weights for WMMA B-fragments:
//   PWd[p*256 + o] = { (Wa-Wb)[o][2p], (Wa-Wb)[o][2p+1] }
//   PWb[p*256 + o] = {      Wb [o][2p],      Wb [o][2p+1] }
// W is [256][512] row-major: Wa = W[:, 0:256], Wb = W[:, 256:512].
// ---------------------------------------------------------------------------
__global__ __launch_bounds__(256) void prep_weights_kernel(
    const float* __restrict__ W, float2* __restrict__ PWd, float2* __restrict__ PWb)
{
    int idx = blockIdx.x * 256 + threadIdx.x;     // 128 * 256 pair-slots
    if (idx >= 128 * 256) return;
    int p = idx >> 8;                             // K pair index
    int o = idx & 255;                            // output channel
    int k = p << 1;
    float a0 = W[o * 512 + k];
    float a1 = W[o * 512 + k + 1];
    float b0 = W[o * 512 + 256 + k];
    float b1 = W[o * 512 + 256 + k + 1];
    PWd[idx] = make_float2(a0 - b0, a1 - b1);
    PWb[idx] = make_float2(b0, b1);
}

__global__ void zero_kernel(float* __restrict__ p, size_t n)
{
    size_t i = (size_t)blockIdx.x * blockDim.x + threadIdx.x;
    size_t stride = (size_t)gridDim.x * blockDim.x;
    for (; i < n; i += stride) p[i] = 0.0f;
}

// ---------------------------------------------------------------------------
// Y[N,256] = X[N,256] @ Wpacked^T using V_WMMA_F32_16X16X4_F32.
// Block = 256 threads (8 waves). Block computes a 16-row x 128-col tile strip
// (blockIdx.y selects column half). X strip staged once in LDS; each wave owns
// one 16x16 tile and runs the K=256 loop in steps of 4 (64 WMMAs).
//
// Fragment layout (wave32, f32):
//   A 16x4 : lanes 0-15 -> rows, VGPR{0,1} = K{0,1}; lanes 16-31 = K{2,3}
//   B 4x16 : lanes 0-15 -> cols, VGPR{0,1} = K{0,1}; lanes 16-31 = K{2,3}
//   C/D    : VGPR r: lanes 0-15 = row r, lanes 16-31 = row r+8
// ---------------------------------------------------------------------------
__global__ __launch_bounds__(256) void gemm_wmma_f32_kernel(
    const float*  __restrict__ X,
    const float2* __restrict__ PW,    // [128][256] packed K-pairs
    float*        __restrict__ Y)
{
    __shared__ float lds_x[16 * LDS_STRIDE];

    const int t  = threadIdx.x;
    const int m0 = blockIdx.x << 4;              // 16-row strip (50000/16 = 3125)

    // Cooperative load of X[m0:m0+16, 0:256] -> LDS (each thread: 4 x float4)
    {
        int row = t >> 4;                        // 0..15
        int col = (t & 15) << 4;                 // 0,16,...,240
        const float4* src4 = reinterpret_cast<const float4*>(
            X + (size_t)(m0 + row) * CCH + col);
        float* dst = &lds_x[row * LDS_STRIDE + col];
#pragma unroll
        for (int i = 0; i < 4; ++i) {
            float4 v = src4[i];
            dst[i * 4 + 0] = v.x; dst[i * 4 + 1] = v.y;
            dst[i * 4 + 2] = v.z; dst[i * 4 + 3] = v.w;
        }
    }
    __syncthreads();

    const int wave  = t >> 5;                    // 0..7 -> 16-col tile in strip
    const int lane  = t & 31;
    const int lrow  = lane & 15;                 // row (A) / col (B) within tile
    const int khalf = lane >> 4;                 // K sub-pair select
    const int n0    = (blockIdx.y << 7) + (wave << 4);

    const float* xrow = &lds_x[lrow * LDS_STRIDE];
    v8f acc = {};

#pragma unroll 8
    for (int k0 = 0; k0 < CCH; k0 += 4) {
        int kk = k0 + (khalf << 1);
        v2f a;
        a.x = xrow[kk];
        a.y = xrow[kk + 1];
        float2 bw = PW[(size_t)(kk >> 1) * CCH + n0 + lrow];
        v2f b;
        b.x = bw.x;
        b.y = bw.y;
        acc = __builtin_amdgcn_wmma_f32_16x16x4_f32(
            /*neg_a=*/false, a, /*neg_b=*/false, b,
            /*c_mod=*/(short)0, acc, /*reuse_a=*/false, /*reuse_b=*/false);
    }

#pragma unroll
    for (int r = 0; r < 8; ++r) {
        int row = m0 + r + (khalf << 3);
        Y[(size_t)row * CCH + n0 + lrow] = acc[r];
    }
}

// ---------------------------------------------------------------------------
// Edge stages: thread t = channel, block walks EPB edges.
//   val = relu(P[dst][c] + Q[src][c] + b[c])
// max pass : global_atomic_max_u32 on bit pattern (valid: val >= 0, out init 0)
// sum pass : global_atomic_add_f32 (no return)
// ---------------------------------------------------------------------------
#define EPB 8

__global__ __launch_bounds__(256) void edge_max_kernel(
    const long long* __restrict__ ei,   // [2][E]: row0 = src, row1 = dst
    const float* __restrict__ P, const float* __restrict__ Q,
    const float* __restrict__ bias, float* __restrict__ out, int E)
{
    const int c  = threadIdx.x;
    const float bc = bias[c];
    const int e0 = blockIdx.x * EPB;
#pragma unroll
    for (int i = 0; i < EPB; ++i) {
        int e = e0 + i;
        if (e >= E) return;
        long long s = ei[e];
        long long d = ei[(size_t)E + e];
        float val = P[(size_t)d * CCH + c] + Q[(size_t)s * CCH + c] + bc;
        if (val > 0.0f) {
            atomicMax(reinterpret_cast<unsigned int*>(out) + (size_t)d * CCH + c,
                      __float_as_uint(val));
        }
    }
}

__global__ __launch_bounds__(256) void edge_sum_kernel(
    const long long* __restrict__ ei,
    const float* __restrict__ P, const float* __restrict__ Q,
    const float* __restrict__ bias, float* __restrict__ out, int E)
{
    const int c  = threadIdx.x;
    const float bc = bias[c];
    const int e0 = blockIdx.x * EPB;
#pragma unroll
    for (int i = 0; i < EPB; ++i) {
        int e = e0 + i;
        if (e >= E) return;
        long long s = ei[e];
        long long d = ei[(size_t)E + e];
        float val = P[(size_t)d * CCH + c] + Q[(size_t)s * CCH + c] + bc;
        if (val > 0.0f) {
            atomicAdd(out + (size_t)d * CCH + c, val);
        }
    }
}

// ---------------------------------------------------------------------------
extern "C" void kernel_launch(void* const* d_in, const int* in_sizes, int n_in,
                              void* d_out, int out_size, void* d_ws, size_t ws_size,
                              hipStream_t stream)
{
    const float*     x      = (const float*)d_in[0];
    const long long* ei_tp  = (const long long*)d_in[1];
    const long long* ei_int = (const long long*)d_in[2];
    const float*     W_tp   = (const float*)d_in[3];
    const float*     b_tp   = (const float*)d_in[4];
    const float*     W_int  = (const float*)d_in[5];
    const float*     b_int  = (const float*)d_in[6];
    float* out = (float*)d_out;

    const int E = in_sizes[1] / 2;               // 500000

    // Workspace layout
    char* ws = (char*)d_ws;
    float2* PWd_tp  = (float2*)(ws);                      // 256 KB each
    float2* PWb_tp  = (float2*)(ws + (256u << 10));
    float2* PWd_int = (float2*)(ws + (512u << 10));
    float2* PWb_int = (float2*)(ws + (768u << 10));
    float*  Pbuf    = (float*)(ws + (1u << 20));          // 51.2 MB
    float*  Qbuf    = Pbuf + (size_t)N_NODES * CCH;       // 51.2 MB

    const dim3 gemm_grid(N_NODES / 16, 2);                // 3125 x 2
    const int  edge_blocks = (E + EPB - 1) / EPB;

    // Pack both edge types' weights.
    prep_weights_kernel<<<128, 256, 0, stream>>>(W_tp,  PWd_tp,  PWb_tp);
    prep_weights_kernel<<<128, 256, 0, stream>>>(W_int, PWd_int, PWb_int);
    zero_kernel<<<2048, 256, 0, stream>>>(out, (size_t)N_NODES * CCH);

    // --- 'intersects' edge type (aggr = max) -> writes max values into out
    gemm_wmma_f32_kernel<<<gemm_grid, 256, 0, stream>>>(x, PWd_int, Pbuf);
    gemm_wmma_f32_kernel<<<gemm_grid, 256, 0, stream>>>(x, PWb_int, Qbuf);
    edge_max_kernel<<<edge_blocks, 256, 0, stream>>>(ei_int, Pbuf, Qbuf, b_int, out, E);

    // --- 'temp_previous' edge type (aggr = sum) -> atomicAdd on top (fuses +)
    gemm_wmma_f32_kernel<<<gemm_grid, 256, 0, stream>>>(x, PWd_tp, Pbuf);
    gemm_wmma_f32_kernel<<<gemm_grid, 256, 0, stream>>>(x, PWb_tp, Qbuf);
    edge_sum_kernel<<<edge_blocks, 256, 0, stream>>>(ei_tp, Pbuf, Qbuf, b_tp, out, E);
}